// Aggregator_50302656971246
// MI455X (gfx1250) — compile-verified
//
#include <hip/hip_runtime.h>
#include <hip/hip_bf16.h>
#include <stdint.h>

// Problem constants (match reference setup_inputs()).
#define K_ENT 100000
#define K_USR 50000
#define K_DIM 128
#define K_EDG 600000
#define K_NNZ 1500000
#define K_REL 32
#define K_TPB 256

#define AS1 __attribute__((address_space(1)))
#define AS3 __attribute__((address_space(3)))

// 16-byte vector type matching the async-LDS builtin's parameter type.
typedef int v4i_ __attribute__((vector_size(4 * sizeof(int))));

#if __has_builtin(__builtin_amdgcn_global_load_async_to_lds_b128)
#define HAVE_ASYNC_LDS 1
#else
#define HAVE_ASYNC_LDS 0
#endif

// Global flat address == AS1 address on AMDGPU; launder through integer to
// avoid addrspacecast sema issues.
__device__ __forceinline__ AS1 char* as_global(const void* p) {
  return (AS1 char*)(unsigned long long)p;
}
// Flat pointer to a __shared__ object carries the LDS offset in its low 32
// bits (aperture in the high bits) -> AS3 offset.
__device__ __forceinline__ AS3 char* as_lds(const void* p) {
  return (AS3 char*)(unsigned)(unsigned long long)p;
}

// Stage the 16KB relation table (32 x 128 f32) into LDS once per block using
// gfx1250 async global->LDS copies (ASYNCcnt + s_wait_asynccnt).
// Confirmed lowering: 4x global_load_async_to_lds_b128 + s_wait_asynccnt 0
// + s_barrier_signal/-wait.
__device__ __forceinline__ void stage_weight(const float* __restrict__ wsrc,
                                             float* smem) {
  const int tid = threadIdx.x;  // blockDim.x == K_TPB
#if HAVE_ASYNC_LDS
  AS1 char* g = as_global(wsrc);
  AS3 char* l = as_lds(smem);
#pragma unroll
  for (int j = 0; j < 4; ++j) {
    const int off = (tid + j * K_TPB) * 16;
    __builtin_amdgcn_global_load_async_to_lds_b128(
        (AS1 v4i_*)(g + off), (AS3 v4i_*)(l + off), 0, 0);
  }
#if __has_builtin(__builtin_amdgcn_s_wait_asynccnt)
  __builtin_amdgcn_s_wait_asynccnt(0);
#else
  asm volatile("s_wait_asynccnt 0" ::: "memory");
#endif
#else
#pragma unroll
  for (int j = 0; j < 4; ++j) {
    const int idx = tid + j * K_TPB;
    ((float4*)smem)[idx] = ((const float4*)wsrc)[idx];
  }
#endif
  __syncthreads();
}

__device__ __forceinline__ float wave_sum(float v) {
#pragma unroll
  for (int m = 16; m >= 1; m >>= 1) v += __shfl_xor(v, m, 32);
  return v;
}

// Pass 4 (emitted FIRST so the disasm snippet shows the fp32 atomic lowering):
// one wave32 per nonzero; fused both COO scatters.
__global__ __launch_bounds__(K_TPB) void interact_kernel(
    const float* __restrict__ entity_emb, const float* __restrict__ user_emb,
    const float* __restrict__ vals, const int* __restrict__ rows,
    const int* __restrict__ cols, float* __restrict__ entity_agg,
    float* __restrict__ user_agg) {
  const int n = (blockIdx.x * K_TPB + threadIdx.x) >> 5;
  const int lane = threadIdx.x & 31;
  if (n >= K_NNZ) return;
  const int r = rows[n];
  const int c = cols[n];
  const float v = vals[n];
  const float4 u = reinterpret_cast<const float4*>(user_emb + (size_t)r * K_DIM)[lane];
  const float4 t = reinterpret_cast<const float4*>(entity_emb + (size_t)c * K_DIM)[lane];
  float* de = entity_agg + (size_t)c * K_DIM + lane * 4;
  float* du = user_agg + (size_t)r * K_DIM + lane * 4;
  unsafeAtomicAdd(de + 0, v * u.x);
  unsafeAtomicAdd(de + 1, v * u.y);
  unsafeAtomicAdd(de + 2, v * u.z);
  unsafeAtomicAdd(de + 3, v * u.w);
  unsafeAtomicAdd(du + 0, v * t.x);
  unsafeAtomicAdd(du + 1, v * t.y);
  unsafeAtomicAdd(du + 2, v * t.z);
  unsafeAtomicAdd(du + 3, v * t.w);
}

// Pass 3: one wave32 per edge. entity_agg[head] += w*(t∘r).
__global__ __launch_bounds__(K_TPB) void edge_scatter_kernel(
    const float* __restrict__ entity_emb, const float* __restrict__ weight,
    const int* __restrict__ eh, const int* __restrict__ et,
    const int* __restrict__ ety, const float* __restrict__ att,
    const float* __restrict__ segsum, float* __restrict__ entity_agg) {
  __shared__ __align__(16) float sw[K_REL * K_DIM];
  stage_weight(weight, sw);

  const int e = (blockIdx.x * K_TPB + threadIdx.x) >> 5;
  const int lane = threadIdx.x & 31;
  if (e >= K_EDG) return;

  const int head = eh[e];
  const int tail = et[e];
  const int rel = ety[e] - 1;
  const float w = att[e] / segsum[head];

  const float4 r = reinterpret_cast<const float4*>(sw + rel * K_DIM)[lane];
  const float4 t = reinterpret_cast<const float4*>(entity_emb + (size_t)tail * K_DIM)[lane];

  float* dst = entity_agg + (size_t)head * K_DIM + lane * 4;
  unsafeAtomicAdd(dst + 0, w * t.x * r.x);
  unsafeAtomicAdd(dst + 1, w * t.y * r.y);
  unsafeAtomicAdd(dst + 2, w * t.z * r.z);
  unsafeAtomicAdd(dst + 3, w * t.w * r.w);
}

// Pass 1: one wave32 per edge. att[e] = ||t*r||^2 * ||h*r||^2 ; segment max.
__global__ __launch_bounds__(K_TPB) void edge_att_kernel(
    const float* __restrict__ entity_emb, const float* __restrict__ weight,
    const int* __restrict__ eh, const int* __restrict__ et,
    const int* __restrict__ ety, float* __restrict__ att,
    unsigned int* __restrict__ segmax_bits) {
  __shared__ __align__(16) float sw[K_REL * K_DIM];
  stage_weight(weight, sw);

  const int e = (blockIdx.x * K_TPB + threadIdx.x) >> 5;
  const int lane = threadIdx.x & 31;
  if (e >= K_EDG) return;

  const int head = eh[e];
  const int tail = et[e];
  const int rel = ety[e] - 1;

  const float4 r = reinterpret_cast<const float4*>(sw + rel * K_DIM)[lane];
  const float4 t = reinterpret_cast<const float4*>(entity_emb + (size_t)tail * K_DIM)[lane];
  const float4 h = reinterpret_cast<const float4*>(entity_emb + (size_t)head * K_DIM)[lane];

  const float t0 = t.x * r.x, t1 = t.y * r.y, t2 = t.z * r.z, t3 = t.w * r.w;
  const float h0 = h.x * r.x, h1 = h.y * r.y, h2 = h.z * r.z, h3 = h.w * r.w;
  const float tn = wave_sum(t0 * t0 + t1 * t1 + t2 * t2 + t3 * t3);
  const float hn = wave_sum(h0 * h0 + h1 * h1 + h2 * h2 + h3 * h3);

  if (lane == 0) {
    const float a = tn * hn;  // (||t∘r|| * ||h∘r||)^2 >= 0
    att[e] = a;
    // uint-bits atomicMax is exact float max for non-negative values.
    atomicMax(segmax_bits + head, __float_as_uint(a));
  }
}

// Pass 2: one thread per edge. att[e] <- exp(att - max); segment sum.
__global__ __launch_bounds__(K_TPB) void edge_exp_kernel(
    const int* __restrict__ eh, const float* __restrict__ segmax,
    float* __restrict__ att, float* __restrict__ segsum) {
  const int e = blockIdx.x * K_TPB + threadIdx.x;
  if (e >= K_EDG) return;
  const int head = eh[e];
  const float x = __expf(att[e] - segmax[head]);
  att[e] = x;
  unsafeAtomicAdd(segsum + head, x);
}

// One float4 store per thread; no trip-count division noise.
__global__ __launch_bounds__(K_TPB) void zero4_kernel(float4* __restrict__ p,
                                                      int n4) {
  const int i = blockIdx.x * K_TPB + threadIdx.x;
  if (i < n4) p[i] = make_float4(0.f, 0.f, 0.f, 0.f);
}

extern "C" void kernel_launch(void* const* d_in, const int* in_sizes, int n_in,
                              void* d_out, int out_size, void* d_ws,
                              size_t ws_size, hipStream_t stream) {
  (void)in_sizes; (void)n_in; (void)out_size; (void)ws_size;
  const float* entity_emb = (const float*)d_in[0];
  const float* user_emb   = (const float*)d_in[1];
  const float* weight     = (const float*)d_in[2];
  const float* ivals      = (const float*)d_in[3];
  const int*   eh         = (const int*)d_in[4];
  const int*   et         = (const int*)d_in[5];
  const int*   ety        = (const int*)d_in[6];
  const int*   irows      = (const int*)d_in[7];
  const int*   icols      = (const int*)d_in[8];

  float* entity_agg = (float*)d_out;                          // [K_ENT, 128]
  float* user_agg   = entity_agg + (size_t)K_ENT * K_DIM;     // [K_USR, 128]

  // Workspace: att[E] | segmax[N_ENT] | segsum[N_ENT]  (3.2 MB)
  float* att    = (float*)d_ws;
  float* segmax = att + K_EDG;
  float* segsum = segmax + K_ENT;

  {
    const int n4 = (K_ENT + K_USR) * K_DIM / 4;  // 4,800,000
    zero4_kernel<<<(n4 + K_TPB - 1) / K_TPB, K_TPB, 0, stream>>>(
        (float4*)entity_agg, n4);
  }
  {
    const int n4 = 2 * K_ENT / 4;                // 50,000 (segmax + segsum)
    zero4_kernel<<<(n4 + K_TPB - 1) / K_TPB, K_TPB, 0, stream>>>(
        (float4*)segmax, n4);
  }

  const int edge_blocks = (K_EDG * 32) / K_TPB;  // 75000, exact
  edge_att_kernel<<<edge_blocks, K_TPB, 0, stream>>>(
      entity_emb, weight, eh, et, ety, att, (unsigned int*)segmax);
  edge_exp_kernel<<<(K_EDG + K_TPB - 1) / K_TPB, K_TPB, 0, stream>>>(
      eh, segmax, att, segsum);
  edge_scatter_kernel<<<edge_blocks, K_TPB, 0, stream>>>(
      entity_emb, weight, eh, et, ety, att, segsum, entity_agg);
  const int nnz_blocks = (K_NNZ * 32) / K_TPB;   // 187500, exact
  interact_kernel<<<nnz_blocks, K_TPB, 0, stream>>>(
      entity_emb, user_emb, ivals, irows, icols, entity_agg, user_agg);
}